// NeuralRenderer_66099546685909
// MI455X (gfx1250) — compile-verified
//
#include <hip/hip_runtime.h>
#include <cstdint>

constexpr int   B        = 4;
constexpr int   V        = 778;
constexpr int   F        = 1538;
constexpr int   F2       = 2 * F;          // 3076 (front + back faces)
constexpr int   TRI_CHUNK= 32;
constexpr int   NPAD     = 3104;           // F2 padded to multiple of 32 (97 chunks)
constexpr int   NCHUNK   = NPAD / TRI_CHUNK;
constexpr int   RASTK    = 256;
constexpr float FARZ     = 10.0f;
constexpr float EPSZ     = 1e-9f;

// ---------------- CDNA5 async global->LDS helpers (gfx1250) ----------------
#if defined(__HIP_DEVICE_COMPILE__)
#if __has_builtin(__builtin_amdgcn_global_load_async_to_lds_b64)
#define ATHENA_HAVE_ASYNC_LDS 1
#endif
#endif

#if defined(ATHENA_HAVE_ASYNC_LDS)
typedef int athena_v2i __attribute__((vector_size(2 * sizeof(int))));
typedef athena_v2i __attribute__((address_space(1)))* athena_gptr_v2i;
typedef athena_v2i __attribute__((address_space(3)))* athena_lptr_v2i;
#endif

__device__ __forceinline__ void async_g2l_b64(const float* __restrict__ g, float* l) {
#if defined(ATHENA_HAVE_ASYNC_LDS)
  // per-lane 8B DMA from the global address g to the LDS address l; tracked by ASYNCcnt
  __builtin_amdgcn_global_load_async_to_lds_b64(
      (athena_gptr_v2i)(athena_v2i*)(void*)(g),
      (athena_lptr_v2i)(athena_v2i*)(void*)(l), 0, 0);
#else
  // synchronous fallback — still correct with the surrounding barriers
  float2 v = *(const float2*)g;
  *(float2*)l = v;
#endif
}

__device__ __forceinline__ void wait_async_zero() {
#if defined(__HIP_DEVICE_COMPILE__)
#if __has_builtin(__builtin_amdgcn_s_wait_asynccnt)
  __builtin_amdgcn_s_wait_asynccnt(0);
#else
  asm volatile("s_wait_asynccnt 0" ::: "memory");
#endif
#endif
}

// ---------------- kernels ---------------------------------------------------

__global__ void k_init(unsigned* __restrict__ cells) {
  int i = threadIdx.x;
  if (i < B) {
    cells[i]     = 0u;                       // running max of masked depth (>=0)
    cells[B + i] = __float_as_uint(FARZ);    // running min of depth (<= FAR)
  }
}

// ---- WMMA rotate pass: cam[b,v,:] = einsum('vj,ij->vi', verts, R[b]) + t[b]
// V_WMMA_F32_16X16X4_F32:  D[16x16] = A[16x4] x B[4x16] + C
//   A row M = vertex (x,y,z,1);  B col N (<3) = {R[N][0],R[N][1],R[N][2],t[N]}
typedef float athena_v2f __attribute__((ext_vector_type(2)));
typedef float athena_v8f __attribute__((ext_vector_type(8)));

__global__ __launch_bounds__(256)
void k_rotate(const float* __restrict__ verts,
              const float* __restrict__ Rm,
              const float* __restrict__ tm,
              float* __restrict__ cam) {
  constexpr int TILES_PER_B = (V + 15) / 16;   // 49 wave-tiles per batch
  const int lane = threadIdx.x & 31;
  const int wave = blockIdx.x * (blockDim.x >> 5) + (threadIdx.x >> 5);
  if (wave >= B * TILES_PER_B) return;         // wave-uniform guard (EXEC stays full)
  const int b  = wave / TILES_PER_B;
  const int v0 = (wave - b * TILES_PER_B) * 16;
  const int n  = lane & 15;
  const bool hi = lane >= 16;

  // A operand: lanes 0-15 -> {K0=x, K1=y}; lanes 16-31 -> {K2=z, K3=1}
  int vclamp = v0 + n; if (vclamp > V - 1) vclamp = V - 1;   // pad rows: clamp
  const float* vp = verts + (size_t)(b * V + vclamp) * 3;
  athena_v2f a;
  if (!hi) { a.x = vp[0]; a.y = vp[1]; }
  else     { a.x = vp[2]; a.y = 1.0f; }

  // B operand: lanes 0-15 -> rows {K0,K1}; lanes 16-31 -> rows {K2,K3}; col = n
  const float* R = Rm + b * 9;
  const float* t = tm + b * 3;
  athena_v2f bm;
  if (!hi) { bm.x = (n < 3) ? R[n * 3 + 0] : 0.0f;
             bm.y = (n < 3) ? R[n * 3 + 1] : 0.0f; }
  else     { bm.x = (n < 3) ? R[n * 3 + 2] : 0.0f;
             bm.y = (n < 3) ? t[n]         : 0.0f; }

  athena_v8f c = {0.f, 0.f, 0.f, 0.f, 0.f, 0.f, 0.f, 0.f};
  athena_v8f d = __builtin_amdgcn_wmma_f32_16x16x4_f32(
      /*neg_a=*/false, a, /*neg_b=*/false, bm,
      /*c_mod=*/(short)0, c, /*reuse_a=*/false, /*reuse_b=*/false);

  // D: lane = column (component n), VGPR r = row r (lanes>=16: row r+8)
  if (n < 3) {
    const int mbase = hi ? 8 : 0;
#pragma unroll
    for (int r = 0; r < 8; ++r) {
      const int m = v0 + mbase + r;
      if (m < V) cam[(size_t)(b * V + m) * 3 + n] = d[r];
    }
  }
}

// Distortion + intrinsics + NDC flip (camera-space verts come from k_rotate)
__global__ void k_project(const float* __restrict__ cam,
                          const float* __restrict__ Km,
                          const float* __restrict__ dist,
                          float* __restrict__ uvz) {
  int gid = blockIdx.x * blockDim.x + threadIdx.x;
  if (gid >= B * V) return;
  int b = gid / V, i = gid - b * V;
  const float* cp = cam + (size_t)(b * V + i) * 3;
  float x = cp[0], y = cp[1], z = cp[2];
  float zi = z + EPSZ;
  float x_ = x / zi;
  float y_ = y / zi;
  const float* d = dist + b * 5;
  float k1 = d[0], k2 = d[1], p1 = d[2], p2 = d[3], k3 = d[4];
  float r2 = x_*x_ + y_*y_;
  float r4 = r2 * r2;
  float r6 = r4 * r2;
  float radial = 1.0f + k1*r2 + k2*r4 + k3*r6;
  float x__ = x_*radial + 2.0f*p1*x_*y_ + p2*(r2 + 2.0f*x_*x_);
  float y__ = y_*radial + p1*(r2 + 2.0f*y_*y_) + 2.0f*p2*x_*y_;
  const float* K = Km + b * 9;
  float u  = K[0]*x__ + K[1]*y__ + K[2];
  float vv = K[3]*x__ + K[4]*y__ + K[5];
  vv = 256.0f - vv;
  u  = 2.0f * (u  - 128.0f) / 256.0f;
  vv = 2.0f * (vv - 128.0f) / 256.0f;
  float* o = uvz + (size_t)(b * V + i) * 3;
  o[0] = u; o[1] = vv; o[2] = z;
}

// Per-triangle setup: edge coeffs, den_s, 1/zs(z), validity flag, eps-padded bbox.
// Record = 16 floats (64B): {e0,e1,e2,e3}{cx,cy,den_s,flag}{raz,rbz,rcz,0}{x0,x1,y0,y1}
__global__ void k_setup(const float* __restrict__ uvz,
                        const int* __restrict__ faces,
                        float* __restrict__ tris) {
  int gid = blockIdx.x * blockDim.x + threadIdx.x;
  if (gid >= B * NPAD) return;
  int b = gid / NPAD, f2 = gid - b * NPAD;
  float4 r0, r1, r2v, r3;
  if (f2 < F2) {
    int i0, i1, i2;
    if (f2 < F) {
      const int* fp = faces + (size_t)(b * F + f2) * 3;
      i0 = fp[0]; i1 = fp[1]; i2 = fp[2];
    } else {                                   // faces[:, :, ::-1]
      const int* fp = faces + (size_t)(b * F + (f2 - F)) * 3;
      i0 = fp[2]; i1 = fp[1]; i2 = fp[0];
    }
    const float* A  = uvz + (size_t)(b * V + i0) * 3;
    const float* Bv = uvz + (size_t)(b * V + i1) * 3;
    const float* Cv = uvz + (size_t)(b * V + i2) * 3;
    float ax = A[0],  ay = A[1],  az = A[2];
    float bx = Bv[0], by = Bv[1], bz = Bv[2];
    float cx = Cv[0], cy = Cv[1], cz = Cv[2];
    float den = (bx - ax) * (cy - ay) - (by - ay) * (cx - ax);
    bool  ok  = fabsf(den) > 1e-8f;
    float den_s = ok ? den : 1.0f;
    bool  zok = (az > 1e-8f) && (bz > 1e-8f) && (cz > 1e-8f);
    bool  flag = ok && zok;
    float zsa = (az > 1e-8f) ? az : 1.0f;
    float zsb = (bz > 1e-8f) ? bz : 1.0f;
    float zsc = (cz > 1e-8f) ? cz : 1.0f;
    r0  = make_float4(by - cy, cx - bx, cy - ay, ax - cx);
    r1  = make_float4(cx, cy, den_s, flag ? 1.0f : 0.0f);
    r2v = make_float4(1.0f / zsa, 1.0f / zsb, 1.0f / zsc, 0.0f);
    if (flag) {
      const float TOL = 1e-4f;  // << pixel pitch (0.0078), >> fp32 ulp
      float x0 = fminf(fminf(ax, bx), cx) - TOL;
      float x1 = fmaxf(fmaxf(ax, bx), cx) + TOL;
      float y0 = fminf(fminf(ay, by), cy) - TOL;
      float y1 = fmaxf(fmaxf(ay, by), cy) + TOL;
      r3 = make_float4(x0, x1, y0, y1);
    } else {
      r3 = make_float4(2.0e9f, -2.0e9f, 2.0e9f, -2.0e9f);  // empty bbox -> culled
    }
  } else {  // zero-padded triangles in the reference scan: never valid
    r0  = make_float4(0.f, 0.f, 0.f, 0.f);
    r1  = make_float4(0.f, 0.f, 1.0f, 0.f);
    r2v = make_float4(0.f, 0.f, 0.f, 0.f);
    r3  = make_float4(2.0e9f, -2.0e9f, 2.0e9f, -2.0e9f);
  }
  float4* o = (float4*)(tris + (size_t)gid * 16);
  o[0] = r0; o[1] = r1; o[2] = r2v; o[3] = r3;
}

// Rasterize: one 16x16 pixel tile per block; triangle stream double-buffered
// through LDS via async global->LDS DMA; bbox-vs-tile culling; z-min + cover.
__global__ __launch_bounds__(256)
void k_raster(const float* __restrict__ tris,
              float* __restrict__ zdep,      // dep region of d_out (z-buffer now)
              float* __restrict__ maskout,   // mask region of d_out
              unsigned* __restrict__ cells) {
  __shared__ float sbuf[2][TRI_CHUNK * 16];  // 2 x 2KB staging buffers
  const int tx = threadIdx.x, ty = threadIdx.y;
  const int tid = ty * 16 + tx;
  const int b  = blockIdx.z;
  const int px = blockIdx.x * 16 + tx;
  const int py = blockIdx.y * 16 + ty;
  const float X = 2.0f * ((float)px + 0.5f) / 256.0f - 1.0f;
  const float Y = 2.0f * ((float)py + 0.5f) / 256.0f - 1.0f;
  const float tXmin = 2.0f * ((float)(blockIdx.x * 16)      + 0.5f) / 256.0f - 1.0f;
  const float tXmax = 2.0f * ((float)(blockIdx.x * 16 + 15) + 0.5f) / 256.0f - 1.0f;
  const float tYmin = 2.0f * ((float)(blockIdx.y * 16)      + 0.5f) / 256.0f - 1.0f;
  const float tYmax = 2.0f * ((float)(blockIdx.y * 16 + 15) + 0.5f) / 256.0f - 1.0f;

  const float* tb = tris + (size_t)b * NPAD * 16;
  float zmin = FARZ;
  bool  cov  = false;

  // prologue: stage chunk 0 (each of the 256 lanes DMAs 8 bytes)
  async_g2l_b64(tb + tid * 2, &sbuf[0][tid * 2]);

  for (int c = 0; c < NCHUNK; ++c) {
    wait_async_zero();      // my wave's slices of chunk c have landed in LDS
    __syncthreads();        // everyone's slices landed; prev buffer free
    if (c + 1 < NCHUNK) {
      async_g2l_b64(tb + (size_t)(c + 1) * (TRI_CHUNK * 16) + tid * 2,
                    &sbuf[(c + 1) & 1][tid * 2]);
    }
    const float* sb = sbuf[c & 1];
#pragma unroll 4
    for (int t = 0; t < TRI_CHUNK; ++t) {
      const float4 bb = *(const float4*)(sb + t * 16 + 12);
      if (bb.y < tXmin || bb.x > tXmax || bb.w < tYmin || bb.z > tYmax) continue;
      const float4 e  = *(const float4*)(sb + t * 16);
      const float4 g  = *(const float4*)(sb + t * 16 + 4);
      const float4 rz = *(const float4*)(sb + t * 16 + 8);
      float dx = X - g.x, dy = Y - g.y;
      float w0 = (e.x * dx + e.y * dy) / g.z;   // / den_s, as in the reference
      float w1 = (e.z * dx + e.w * dy) / g.z;
      float w2 = 1.0f - w0 - w1;
      float inv = w0 * rz.x + w1 * rz.y + w2 * rz.z;
      if ((g.w != 0.0f) & (w0 >= 0.0f) & (w1 >= 0.0f) & (w2 >= 0.0f) & (inv > 1e-8f)) {
        float zp = 1.0f / inv;
        zmin = fminf(zmin, zp);
        cov  = true;
      }
    }
    // next iteration's top barrier orders buffer reuse
  }

  const size_t pix = (size_t)b * (RASTK * RASTK) + (size_t)py * RASTK + px;
  zdep[pix]    = zmin;                 // raw z-buffer, normalized by k_normalize
  maskout[pix] = cov ? 1.0f : 0.0f;

  // block-level reduction for per-image max(masked z) / min(z)
  __syncthreads();                     // done reading sbuf; reuse as scratch
  float noback = (zmin == FARZ) ? 0.0f : zmin;   // (1 - (img==FAR)) * img
  sbuf[0][tid]       = noback;
  sbuf[0][256 + tid] = zmin;
  __syncthreads();
  for (int s = 128; s > 0; s >>= 1) {
    if (tid < s) {
      sbuf[0][tid]       = fmaxf(sbuf[0][tid],       sbuf[0][tid + s]);
      sbuf[0][256 + tid] = fminf(sbuf[0][256 + tid], sbuf[0][256 + tid + s]);
    }
    __syncthreads();
  }
  if (tid == 0) {
    // all values >= 0 -> IEEE float order == uint order
    atomicMax(&cells[b],     __float_as_uint(sbuf[0][0]));
    atomicMin(&cells[B + b], __float_as_uint(sbuf[0][256]));
  }
}

__global__ void k_normalize(float* __restrict__ dep,
                            const unsigned* __restrict__ cells) {
  int gid = blockIdx.x * blockDim.x + threadIdx.x;
  if (gid >= B * RASTK * RASTK) return;
  int b = gid >> 16;
  float mx = __uint_as_float(cells[b]);
  float mn = __uint_as_float(cells[B + b]);
  float z  = dep[gid];
  float nd = (mx - z) / (mx - mn + 0.0001f);
  nd = fminf(fmaxf(nd, 0.0f), 1.0f);
  dep[gid] = nd;
}

// ---------------- launcher --------------------------------------------------

extern "C" void kernel_launch(void* const* d_in, const int* in_sizes, int n_in,
                              void* d_out, int out_size, void* d_ws, size_t ws_size,
                              hipStream_t stream) {
  (void)in_sizes; (void)n_in; (void)out_size; (void)ws_size;
  const float* verts = (const float*)d_in[0];
  const int*   faces = (const int*)d_in[1];
  const float* Km    = (const float*)d_in[2];
  const float* Rm    = (const float*)d_in[3];
  const float* tm    = (const float*)d_in[4];
  const float* dist  = (const float*)d_in[5];

  float* dep  = (float*)d_out;                          // (B,256,256)
  float* mask = dep + (size_t)B * RASTK * RASTK;        // (B,256,256)

  float* ws   = (float*)d_ws;
  float* uvz  = ws;                                     // B*V*3 = 9336 floats
  float* cam  = ws + 9344;                              // B*V*3 = 9336 floats
  float* tris = ws + 18688;                             // B*NPAD*16 = 198656 floats
  unsigned* cells = (unsigned*)(ws + 18688 + (size_t)B * NPAD * 16);  // 8 uints

  constexpr int NWAVES = B * ((V + 15) / 16);           // 196 wave-tiles
  k_init<<<1, 32, 0, stream>>>(cells);
  k_rotate<<<(NWAVES + 7) / 8, 256, 0, stream>>>(verts, Rm, tm, cam);
  k_project<<<(B * V + 255) / 256, 256, 0, stream>>>(cam, Km, dist, uvz);
  k_setup<<<(B * NPAD + 255) / 256, 256, 0, stream>>>(uvz, faces, tris);
  k_raster<<<dim3(16, 16, B), dim3(16, 16, 1), 0, stream>>>(tris, dep, mask, cells);
  k_normalize<<<(B * RASTK * RASTK + 255) / 256, 256, 0, stream>>>(dep, cells);
}